// Scores_2473901163258
// MI455X (gfx1250) — compile-verified
//
#include <hip/hip_runtime.h>

// fp32 WMMA fragments for V_WMMA_F32_16X16X4_F32:
//   A (16x4):  2 VGPRs/lane. lanes 0-15: M=lane, K={0,1}; lanes 16-31: M=lane-16, K={2,3}
//   B (4x16):  2 VGPRs/lane. lanes 0-15: N=lane, K={0,1}; lanes 16-31: N=lane-16, K={2,3}
//   C/D (16x16): 8 VGPRs. VGPR r, lanes 0-15 -> row r, col lane; lanes 16-31 -> row r+8, col lane-16
typedef __attribute__((ext_vector_type(2))) float v2f;
typedef __attribute__((ext_vector_type(8))) float v8f;

#define D_VIS 512
#define D_SEM 28
#define WAVES_PER_WG 8
#define ROWS_PER_WAVE 32                    // two 16-row WMMA tiles per wave
#define ROWS_PER_WG (WAVES_PER_WG * ROWS_PER_WAVE)   // 256 rows per block
#define KSTEPS (D_VIS / 4)                  // 128 WMMA k-steps

__global__ __launch_bounds__(256, 1)
void scores_wmma_kernel(const float* __restrict__ thetas,
                        const float* __restrict__ phi,
                        const float* __restrict__ W,
                        float* __restrict__ out, int N)
{
    // Swizzled W fragments: [step][half][lane] -> (W[k0][j], W[k0+1][j]); 64 KB
    __shared__ v2f ldsW[KSTEPS * 2 * 32];

    const int tid  = threadIdx.x;
    const int lane = tid & 31;
    const int wave = tid >> 5;

    // Cooperatively build B fragments in LDS (W is 56 KB, stays hot in L2 across blocks).
    for (int e = tid; e < KSTEPS * 2 * 32; e += 256) {
        const int s  = e >> 6;          // k-step
        const int h  = (e >> 5) & 1;    // column half (cols 0-15 / 16-31)
        const int L  = e & 31;          // lane slot
        const int k0 = s * 4 + ((L >> 4) << 1);
        const int j  = h * 16 + (L & 15);
        v2f v = {0.0f, 0.0f};
        if (j < D_SEM) {                // zero-pad cols 28-31
            v.x = W[k0 * D_SEM + j];
            v.y = W[(k0 + 1) * D_SEM + j];
        }
        ldsW[e] = v;
    }
    __syncthreads();

    const int row0 = blockIdx.x * ROWS_PER_WG + wave * ROWS_PER_WAVE;

    // Per-lane A pointers for the two 16-row tiles
    const float* aptr0 = thetas + (size_t)(row0 + (lane & 15)) * D_VIS + ((lane >> 4) << 1);
    const float* aptr1 = aptr0 + 16 * D_VIS;
    const v2f*   bptr  = ldsW + lane;

    v8f acc00 = {0.f,0.f,0.f,0.f,0.f,0.f,0.f,0.f};  // tile0, cols 0-15
    v8f acc01 = {0.f,0.f,0.f,0.f,0.f,0.f,0.f,0.f};  // tile0, cols 16-31
    v8f acc10 = {0.f,0.f,0.f,0.f,0.f,0.f,0.f,0.f};  // tile1, cols 0-15
    v8f acc11 = {0.f,0.f,0.f,0.f,0.f,0.f,0.f,0.f};  // tile1, cols 16-31

#pragma unroll 4
    for (int s = 0; s < KSTEPS; ++s) {
        v2f a0 = *(const v2f*)(aptr0 + 4 * s);    // global_load_b64
        v2f a1 = *(const v2f*)(aptr1 + 4 * s);
        v2f b0 = bptr[(s * 2 + 0) * 32];          // ds_load_b64, conflict-free
        v2f b1 = bptr[(s * 2 + 1) * 32];
        acc00 = __builtin_amdgcn_wmma_f32_16x16x4_f32(false, a0, false, b0,
                                                      (short)0, acc00, false, false);
        acc01 = __builtin_amdgcn_wmma_f32_16x16x4_f32(false, a0, false, b1,
                                                      (short)0, acc01, false, false);
        acc10 = __builtin_amdgcn_wmma_f32_16x16x4_f32(false, a1, false, b0,
                                                      (short)0, acc10, false, false);
        acc11 = __builtin_amdgcn_wmma_f32_16x16x4_f32(false, a1, false, b1,
                                                      (short)0, acc11, false, false);
    }

    // Outputs: scores = out[0..N), phi_samples = out[N ..) as [n*28 + j]
    float* scores = out;
    float* psamp  = out + (size_t)N;

    const int j = lane & 15;
    const float ph0 = phi[j];                                   // j < 16 < 28 always valid
    const float ph1 = (j < D_SEM - 16) ? phi[16 + j] : 0.0f;    // cols 16..27 only

#pragma unroll
    for (int t = 0; t < 2; ++t) {
        const v8f accA = t ? acc10 : acc00;
        const v8f accB = t ? acc11 : acc01;
        const int tbase = row0 + t * 16;

#pragma unroll
        for (int r = 0; r < 8; ++r) {
            const int m   = r + ((lane >> 4) << 3);   // row within 16-row tile
            const int row = tbase + m;
            const float o0 = accA[r];
            const float o1 = accB[r];

            // phi_samples: each lane half writes 64 consecutive bytes per row
            psamp[(size_t)row * D_SEM + j] = o0;
            if (j < D_SEM - 16)
                psamp[(size_t)row * D_SEM + 16 + j] = o1;

            // scores: dot with phi, reduce across the 16 column lanes
            float part = o0 * ph0 + o1 * ph1;
            part += __shfl_xor(part, 1, 16);
            part += __shfl_xor(part, 2, 16);
            part += __shfl_xor(part, 4, 16);
            part += __shfl_xor(part, 8, 16);
            if (j == 0)
                scores[row] = part;
        }
    }
}

extern "C" void kernel_launch(void* const* d_in, const int* in_sizes, int n_in,
                              void* d_out, int out_size, void* d_ws, size_t ws_size,
                              hipStream_t stream) {
    const float* thetas = (const float*)d_in[0];   // (N, 512) f32
    const float* phi    = (const float*)d_in[1];   // (28, 1)  f32
    const float* W      = (const float*)d_in[2];   // (512, 28) f32
    float* out = (float*)d_out;                    // N scores ++ N*28 phi_samples

    const int N = in_sizes[0] / D_VIS;             // 131072
    const int blocks = N / ROWS_PER_WG;            // 512
    scores_wmma_kernel<<<blocks, 256, 0, stream>>>(thetas, phi, W, out, N);
}